// EquivariantProductBasisBlock_66056597013084
// MI455X (gfx1250) — compile-verified
//
#include <hip/hip_runtime.h>
#include <hip/hip_bf16.h>

// ---------------------------------------------------------------------------
// EquivariantProductBasisBlock for MI455X (gfx1250, wave32)
// B=2048, C=128, S=10, DIN=9, DOUTS=(1,3,5), PATHS=(1,3,8)
//
// out3[b,c,(l,m)] = sum_{ijk} x_i x_j x_k A3[l,s,m,ijk,c]
//                 + sum_{ij}  x_i x_j     A2[l,s,m,ij,c]
//                 + sum_i     x_i         A1[l,s,m,i,c]
// == monomial-basis GEMM:  X[atom, K] @ T[s,c][K, (l,m)]   (K=819 -> padded 1068)
//
// Stage 0: bucket atoms by species (10 buckets).
// Stage 1: build padded per-(s,c) table T (zero rows for K padding, zero cols n>=9).
// Stage 2: v_wmma_f32_16x16x4_f32 monomial GEMM, 2 atom-tiles x 267 K-steps/wave.
// Stage 3: v_wmma_f32_16x16x4_f32 channel-mix GEMM (Wlin) + 1/sqrt(C) + sc.
// ---------------------------------------------------------------------------

typedef float v2f __attribute__((ext_vector_type(2)));
typedef float v8f __attribute__((ext_vector_type(8)));

static constexpr int Bn = 2048, Cn = 128, Sn = 10;
static constexpr int COLS = 1152;          // 128*(1+3+5)
static constexpr int KP = 1068;            // 81*12 (seg3) + 84 (seg2) + 12 (seg1)

// workspace layout (float offsets)
static constexpr size_t TBL_OFF = 0;
static constexpr size_t TBL_SZ  = (size_t)Sn * Cn * KP * 16;      // 21,872,640 fl
static constexpr size_t O_OFF   = TBL_OFF + TBL_SZ;               // out3 [lm][b][c]
static constexpr size_t O_SZ    = (size_t)9 * Bn * Cn;
static constexpr size_t SORT_OFF = O_OFF + O_SZ;                  // ints from here
// ints: [0..9] counts, [16 + s*Bn + j] atom lists

// ---------------------------------------------------------------------------
// Stage 0a/0b: species bucketing
// ---------------------------------------------------------------------------
__global__ void init_counts_kernel(int* __restrict__ wi) {
  if (threadIdx.x < 16) wi[threadIdx.x] = 0;
}

__global__ __launch_bounds__(256) void scatter_kernel(
    const float* __restrict__ node_attrs, int* __restrict__ wi) {
  int b = blockIdx.x * 256 + threadIdx.x;
  if (b >= Bn) return;
  int s = 0;
#pragma unroll
  for (int e = 0; e < Sn; ++e)
    if (node_attrs[(size_t)b * Sn + e] > 0.5f) s = e;
  int pos = atomicAdd(&wi[s], 1);
  wi[16 + s * Bn + pos] = b;
}

// ---------------------------------------------------------------------------
// Stage 1: T[s][c][kp][n], pair-interleaved: float offset within (s,c) block =
//          (kp>>1)*32 + n*2 + (kp&1)   -> B-frag = one b64 per lane per step.
// kp rows: [0,972): g=(i*9+j) groups of 12, k=0..8 real, 9..11 zero
//          [972,1056): t=i*9+j (81 real + 3 zero)
//          [1056,1068): t=i (9 real + 3 zero)
// col n: 0->(l0,m0), 1..3->(l1,m-1), 4..8->(l2,m-4), 9..15 zero
// ---------------------------------------------------------------------------
__global__ __launch_bounds__(256) void precompute_tbl_kernel(
    const float* __restrict__ U1, const float* __restrict__ U2,
    const float* __restrict__ U3, const float* __restrict__ W1,
    const float* __restrict__ W2, const float* __restrict__ W3,
    float* __restrict__ ws)
{
  size_t gid = (size_t)blockIdx.x * blockDim.x + threadIdx.x;
  if (gid >= TBL_SZ) return;

  int n2    = (int)(gid & 31);
  int n     = n2 >> 1;
  int sl    = n2 & 1;
  size_t r  = gid >> 5;
  int kpair = (int)(r % (KP / 2));
  size_t r2 = r / (KP / 2);
  int c     = (int)(r2 & (Cn - 1));
  int s     = (int)(r2 >> 7);
  int kp    = kpair * 2 + sl;

  float v = 0.f;
  if (n < 9) {
    int l = (n == 0) ? 0 : (n < 4 ? 1 : 2);
    int m = (n == 0) ? 0 : (n < 4 ? n - 1 : n - 4);
    if (kp < 972) {
      int g = kp / 12, k = kp % 12;
      if (k < 9) {
        int i = g / 9, j = g % 9;
        const float* u = U3 + ((size_t)(l * 5 + m) * 729 + i * 81 + j * 9 + k) * 8;
        const float* w = W3 + ((size_t)(l * Sn + s) * 8) * Cn + c;
        float acc = 0.f;
#pragma unroll
        for (int p = 0; p < 8; ++p) acc = fmaf(u[p], w[(size_t)p * Cn], acc);
        v = acc;
      }
    } else if (kp < 1056) {
      int t = kp - 972;
      if (t < 81) {
        int i = t / 9, j = t % 9;
        const float* u = U2 + ((size_t)(l * 5 + m) * 81 + i * 9 + j) * 3;
        const float* w = W2 + ((size_t)(l * Sn + s) * 3) * Cn + c;
        float acc = 0.f;
#pragma unroll
        for (int p = 0; p < 3; ++p) acc = fmaf(u[p], w[(size_t)p * Cn], acc);
        v = acc;
      }
    } else {
      int t = kp - 1056;
      if (t < 9)
        v = U1[(size_t)(l * 5 + m) * 9 + t] * W1[(size_t)(l * Sn + s) * Cn + c];
    }
  }
  ws[gid] = v;
}

// ---------------------------------------------------------------------------
// Stage 2: monomial WMMA GEMM.
// Wave = (species s, channel c, 2 atom-tiles of 16 rows).
// A frag (16x4 fp32): lanes 0-15 hold K{kk,kk+1}, lanes 16-31 K{kk+2,kk+3};
// monomials built from registers x[9] with compile-time indices; the lane-half
// K-split is resolved by 8 hoisted v_cndmask selects (not per-group).
// __launch_bounds__(256,4): cap at 256 VGPRs -> no VGPR-MSB traffic, >=4 w/SIMD.
// ---------------------------------------------------------------------------
__global__ __launch_bounds__(256, 4) void monomial_wmma_kernel(
    const float* __restrict__ node_feats, float* __restrict__ ws,
    const int* __restrict__ wi)
{
  const int s   = blockIdx.z;
  const int cnt = wi[s];
  if (cnt == 0) return;                          // uniform exit
  const int ntiles = (cnt + 15) >> 4;
  const int tg = blockIdx.y;                     // group of 2 tiles
  if (tg * 2 >= ntiles) return;                  // uniform exit

  const int wave = threadIdx.x >> 5;
  const int lane = threadIdx.x & 31;
  const int c    = blockIdx.x * 8 + wave;
  const int rlo  = lane & 15;
  const bool hib = lane >= 16;

  const int* list = wi + 16 + s * Bn;
  const int row0 = tg * 32 + rlo;
  const int row1 = row0 + 16;
  const int a0 = list[row0 < cnt ? row0 : cnt - 1];
  const int a1 = list[row1 < cnt ? row1 : cnt - 1];

  float x0[9], x1[9];
#pragma unroll
  for (int k = 0; k < 9; ++k) {
    x0[k] = node_feats[((size_t)a0 * Cn + c) * 9 + k];
    x1[k] = node_feats[((size_t)a1 * Cn + c) * 9 + k];
  }

  // Hoisted k-selects (depend only on lane half, not on the (i,j) group):
  // lanes 0-15 take k in {kb, kb+1}; lanes 16-31 take k in {kb+2, kb+3}.
  const float s0a = hib ? x0[2] : x0[0], s0b = hib ? x0[3] : x0[1];   // kb=0
  const float s0c = hib ? x0[6] : x0[4], s0d = hib ? x0[7] : x0[5];   // kb=4
  const float s1a = hib ? x1[2] : x1[0], s1b = hib ? x1[3] : x1[1];
  const float s1c = hib ? x1[6] : x1[4], s1d = hib ? x1[7] : x1[5];

  // lane-fixed base into the pair-interleaved table
  const float* tb_lane = ws + TBL_OFF + ((size_t)s * Cn + c) * ((size_t)KP * 16)
                       + (size_t)rlo * 2 + (hib ? 32 : 0);

  v8f acc0 = {0.f, 0.f, 0.f, 0.f, 0.f, 0.f, 0.f, 0.f};
  v8f acc1 = {0.f, 0.f, 0.f, 0.f, 0.f, 0.f, 0.f, 0.f};

  auto step = [&](int kk, float m00, float m01, float m10, float m11) {
    v2f bf = *(const v2f*)(tb_lane + (size_t)kk * 16);
    v2f af0; af0.x = m00; af0.y = m01;
    v2f af1; af1.x = m10; af1.y = m11;
    acc0 = __builtin_amdgcn_wmma_f32_16x16x4_f32(false, af0, false, bf,
                                                 (short)0, acc0, false, false);
    acc1 = __builtin_amdgcn_wmma_f32_16x16x4_f32(false, af1, false, bf,
                                                 (short)0, acc1, false, false);
  };

  // ---- seg3: 81 (i,j) groups of 12 k-rows (k=0..8 real, 9..11 zero rows)
#pragma unroll
  for (int g = 0; g < 81; ++g) {
    const int gi = g / 9, gj = g % 9;
    const float p0 = x0[gi] * x0[gj];
    const float p1 = x1[gi] * x1[gj];
    step(g * 12 + 0, p0 * s0a, p0 * s0b, p1 * s1a, p1 * s1b);
    step(g * 12 + 4, p0 * s0c, p0 * s0d, p1 * s1c, p1 * s1d);
    const float q0 = p0 * x0[8];           // rows 8..11: only k=8 real
    const float q1 = p1 * x1[8];
    step(g * 12 + 8, q0, q0, q1, q1);
  }

  // ---- seg2: 84 rows (t = i*9+j; 81 real + 3 zero rows)
#pragma unroll
  for (int t0 = 0; t0 < 84; t0 += 4) {
    const int ta = t0, tb = (t0 + 1 < 81) ? t0 + 1 : 80,
              tc = (t0 + 2 < 81) ? t0 + 2 : 80, td = (t0 + 3 < 81) ? t0 + 3 : 80;
    const float Pa0 = x0[ta / 9] * x0[ta % 9], Pb0 = x0[tb / 9] * x0[tb % 9];
    const float Pc0 = x0[tc / 9] * x0[tc % 9], Pd0 = x0[td / 9] * x0[td % 9];
    const float Pa1 = x1[ta / 9] * x1[ta % 9], Pb1 = x1[tb / 9] * x1[tb % 9];
    const float Pc1 = x1[tc / 9] * x1[tc % 9], Pd1 = x1[td / 9] * x1[td % 9];
    step(972 + t0, hib ? Pc0 : Pa0, hib ? Pd0 : Pb0,
                   hib ? Pc1 : Pa1, hib ? Pd1 : Pb1);
  }

  // ---- seg1: 12 rows (t = i; 9 real + 3 zero rows)
#pragma unroll
  for (int t0 = 0; t0 < 12; t0 += 4) {
    const int ta = (t0 < 9) ? t0 : 8, tb = (t0 + 1 < 9) ? t0 + 1 : 8,
              tc = (t0 + 2 < 9) ? t0 + 2 : 8, td = (t0 + 3 < 9) ? t0 + 3 : 8;
    step(1056 + t0, hib ? x0[tc] : x0[ta], hib ? x0[td] : x0[tb],
                    hib ? x1[tc] : x1[ta], hib ? x1[td] : x1[tb]);
  }

  // ---- store: D frag vgpr r -> row M = r + 8*hi, col n = rlo
  const int n = rlo;
  if (n < 9) {
    float* O = ws + O_OFF + (size_t)n * Bn * Cn;
#pragma unroll
    for (int r = 0; r < 8; ++r) {
      const int Mrow = r + (hib ? 8 : 0);
      const int g0 = tg * 32 + Mrow;
      if (g0 < cnt) O[(size_t)list[g0] * Cn + c] = acc0[r];
      const int g1 = tg * 32 + 16 + Mrow;
      if (g1 < cnt) O[(size_t)list[g1] * Cn + c] = acc1[r];
    }
  }
}

// ---------------------------------------------------------------------------
// Stage 3: y[b,d,(l,m)] = (1/sqrt(C)) * sum_c O[lm][b][c] * Wlin[l][c][d] + sc
// ---------------------------------------------------------------------------
__global__ __launch_bounds__(128) void linear_wmma_kernel(
    const float* __restrict__ ws, const float* __restrict__ Wlin,
    const float* __restrict__ sc, float* __restrict__ out)
{
  const int lane = threadIdx.x & 31;
  const int wave = threadIdx.x >> 5;
  const int lm   = blockIdx.z;                         // 0..8
  const int l    = (lm == 0) ? 0 : (lm < 4 ? 1 : 2);
  const int m    = (lm == 0) ? 0 : (lm < 4 ? lm - 1 : lm - 4);
  const int dl   = (l == 0) ? 1 : (l == 1 ? 3 : 5);
  const int base_col = (l == 0) ? 0 : (l == 1 ? 128 : 512);

  const int b0 = (blockIdx.x * 4 + wave) * 16;
  const int d0 = blockIdx.y * 16;

  const float* __restrict__ A  = ws + O_OFF + (size_t)lm * Bn * Cn;   // [B][C]
  const float* __restrict__ Bm = Wlin + (size_t)l * Cn * Cn;          // [c][d]

  const int row = lane & 15;
  const int hi  = lane >> 4;

  v8f acc = {0.f, 0.f, 0.f, 0.f, 0.f, 0.f, 0.f, 0.f};
  for (int kk = 0; kk < Cn; kk += 4) {
    const float* ap = A + (size_t)(b0 + row) * Cn + kk + 2 * hi;
    v2f af; af.x = ap[0]; af.y = ap[1];
    const float* bp = Bm + (size_t)(kk + 2 * hi) * Cn + d0 + row;
    v2f bf; bf.x = bp[0]; bf.y = bp[Cn];
    acc = __builtin_amdgcn_wmma_f32_16x16x4_f32(
        false, af, false, bf, (short)0, acc, false, false);
  }

  const float inv_sqrt_c = 0.088388347648318447f;      // 1/sqrt(128)
  const int d = d0 + row;
#pragma unroll
  for (int r = 0; r < 8; ++r) {
    const int bb  = b0 + r + 8 * hi;
    const int col = base_col + d * dl + m;
    const size_t oidx = (size_t)bb * COLS + col;
    out[oidx] = fmaf(acc[r], inv_sqrt_c, sc[oidx]);
  }
}

// ---------------------------------------------------------------------------
extern "C" void kernel_launch(void* const* d_in, const int* in_sizes, int n_in,
                              void* d_out, int out_size, void* d_ws, size_t ws_size,
                              hipStream_t stream) {
  (void)in_sizes; (void)n_in; (void)out_size; (void)ws_size;
  const float* node_feats = (const float*)d_in[0];
  const float* node_attrs = (const float*)d_in[1];
  const float* sc         = (const float*)d_in[2];
  const float* U1         = (const float*)d_in[3];
  const float* U2         = (const float*)d_in[4];
  const float* U3         = (const float*)d_in[5];
  const float* W1         = (const float*)d_in[6];
  const float* W2         = (const float*)d_in[7];
  const float* W3         = (const float*)d_in[8];
  const float* Wlin       = (const float*)d_in[9];
  float*       out        = (float*)d_out;
  float*       ws         = (float*)d_ws;
  int*         wi         = (int*)(ws + SORT_OFF);

  // Stage 0: bucket atoms by species
  init_counts_kernel<<<1, 32, 0, stream>>>(wi);
  scatter_kernel<<<Bn / 256, 256, 0, stream>>>(node_attrs, wi);

  // Stage 1: folded monomial table (one thread per element)
  {
    int blocks = (int)((TBL_SZ + 255) / 256);
    precompute_tbl_kernel<<<blocks, 256, 0, stream>>>(U1, U2, U3, W1, W2, W3, ws);
  }

  // Stage 2: monomial WMMA GEMM (over-launched tile dim; empty blocks exit)
  {
    dim3 grid(Cn / 8, 64, Sn);       // (c-groups, tile-pairs, species)
    monomial_wmma_kernel<<<grid, 256, 0, stream>>>(node_feats, ws, wi);
  }

  // Stage 3: WMMA channel-mixing GEMMs + scale + residual
  {
    dim3 grid(Bn / 16 / 4, Cn / 16, 9);
    linear_wmma_kernel<<<grid, 128, 0, stream>>>(ws, Wlin, sc, out);
  }
}